// ImageQueryBranch_8770323219129
// MI455X (gfx1250) — compile-verified
//
#include <hip/hip_runtime.h>
#include <math.h>

typedef __attribute__((ext_vector_type(2))) float v2f;
typedef __attribute__((ext_vector_type(8))) float v8f;

#define C_I   128
#define DVOL  110592      // 48*48*48
#define NBUCK 128         // atomic buckets per (b,c) stat
#define LLEN  20

// ---------------------------------------------------------------------------
// f32 WMMA 16x16x4 helper:  D = A(16x4) * B(4x16) + C
// A frag: lane l -> row (l&15), cols k + 2*(l>>4) + {0,1}
// B frag: lane l -> col (l&15), rows k + 2*(l>>4) + {0,1}
// C/D   : vgpr i -> row m0 + i + 8*(l>>4), col n0 + (l&15)
// ---------------------------------------------------------------------------
__device__ __forceinline__ v8f wmma_f32(v2f a, v2f b, v8f c) {
  return __builtin_amdgcn_wmma_f32_16x16x4_f32(false, a, false, b, (short)0, c,
                                               false, false);
}

// One wave computes a 16(M) x 32(N) output stripe: two 16x16 accumulators.
// W is row-major [128 x 128]; xs is the input tile transposed: xs[n][cin].
__device__ __forceinline__ void gemm_tile_128(const float* __restrict__ Wm, int m0,
                                              const float (*xs)[130], int lane,
                                              v8f& acc0, v8f& acc1) {
  const int lm = lane & 15;
  const int lh = lane >> 4;
  const float* wrow = Wm + (size_t)(m0 + lm) * C_I + 2 * lh;
#pragma unroll
  for (int k = 0; k < C_I; k += 4) {
    v2f a  = *(const v2f*)(wrow + k);
    v2f b0 = *(const v2f*)(&xs[lm][k + 2 * lh]);
    v2f b1 = *(const v2f*)(&xs[lm + 16][k + 2 * lh]);
    acc0 = wmma_f32(a, b0, acc0);
    acc1 = wmma_f32(a, b1, acc1);
  }
}

// ---------------------------------------------------------------------------
// K1: k/v projections (tiny): k[b,c,l] = sum_cp Wk[c,cp]*lang[b,cp,l] + bk[c]
// ---------------------------------------------------------------------------
__global__ __launch_bounds__(256)
void k_kv(const float* __restrict__ lang,
          const float* __restrict__ Wk, const float* __restrict__ bk,
          const float* __restrict__ Wv, const float* __restrict__ bv,
          float* __restrict__ kbuf, float* __restrict__ vbuf) {
  __shared__ float lL[256 * LLEN];
  const int b = blockIdx.x;
  const float* Lg = lang + (size_t)b * 256 * LLEN;
  for (int e = threadIdx.x; e < 256 * LLEN; e += 256) lL[e] = Lg[e];
  __syncthreads();
  for (int idx = threadIdx.x; idx < C_I * LLEN; idx += 256) {
    const int c = idx / LLEN, l = idx % LLEN;
    float ak = bk[c], av = bv[c];
    for (int cp = 0; cp < 256; ++cp) {
      const float x = lL[cp * LLEN + l];
      ak += Wk[c * 256 + cp] * x;
      av += Wv[c * 256 + cp] * x;
    }
    kbuf[b * C_I * LLEN + idx] = ak;
    vbuf[b * C_I * LLEN + idx] = av;
  }
}

// ---------------------------------------------------------------------------
// K2/K5: 1x1-conv GEMM (optionally two weight sets sharing one input tile),
// writes raw output (b,c,p) and accumulates bucketed per-(b,c) sum / sumsq
// for InstanceNorm.  Block = 256 threads = 8 waves, tile = 128 x 32.
// ---------------------------------------------------------------------------
__global__ __launch_bounds__(256)
void k_gemm2(const float* __restrict__ X,
             const float* __restrict__ W1, const float* __restrict__ b1,
             float* __restrict__ O1, float* __restrict__ sum1, float* __restrict__ sq1,
             const float* __restrict__ W2, const float* __restrict__ b2,
             float* __restrict__ O2, float* __restrict__ sum2, float* __restrict__ sq2) {
  __shared__ float xs[32][130];
  __shared__ float bias1[C_I];
  __shared__ float bias2[C_I];
  const int tid    = threadIdx.x;
  const int blk    = blockIdx.x;
  const int bidx   = blk / (DVOL / 32);
  const int p0     = (blk % (DVOL / 32)) * 32;
  const int bucket = blk & (NBUCK - 1);
  const float* Xb = X + (size_t)bidx * C_I * DVOL;

  for (int e = tid; e < C_I * 32; e += 256) {
    const int c = e >> 5, j = e & 31;
    xs[j][c] = Xb[(size_t)c * DVOL + p0 + j];
  }
  if (tid < C_I) {
    bias1[tid] = b1[tid];
    if (W2) bias2[tid] = b2[tid];
  }
  __syncthreads();

  const int wave = tid >> 5, lane = tid & 31;
  const int lm = lane & 15, lh = lane >> 4;
  const int m0 = wave * 16;

  {
    v8f a0 = {0, 0, 0, 0, 0, 0, 0, 0}, a1 = {0, 0, 0, 0, 0, 0, 0, 0};
    gemm_tile_128(W1, m0, xs, lane, a0, a1);
    float* Ob = O1 + (size_t)bidx * C_I * DVOL;
#pragma unroll
    for (int i = 0; i < 8; ++i) {
      const int c = m0 + i + 8 * lh;
      const float v0 = a0[i] + bias1[c];
      const float v1 = a1[i] + bias1[c];
      Ob[(size_t)c * DVOL + p0 + lm]      = v0;
      Ob[(size_t)c * DVOL + p0 + 16 + lm] = v1;
      float s = v0 + v1, q = v0 * v0 + v1 * v1;
      s += __shfl_xor(s, 1, 32); s += __shfl_xor(s, 2, 32);
      s += __shfl_xor(s, 4, 32); s += __shfl_xor(s, 8, 32);
      q += __shfl_xor(q, 1, 32); q += __shfl_xor(q, 2, 32);
      q += __shfl_xor(q, 4, 32); q += __shfl_xor(q, 8, 32);
      if (lm == 0) {
        atomicAdd(&sum1[(size_t)(bidx * C_I + c) * NBUCK + bucket], s);
        atomicAdd(&sq1 [(size_t)(bidx * C_I + c) * NBUCK + bucket], q);
      }
    }
  }
  if (W2) {
    v8f a0 = {0, 0, 0, 0, 0, 0, 0, 0}, a1 = {0, 0, 0, 0, 0, 0, 0, 0};
    gemm_tile_128(W2, m0, xs, lane, a0, a1);
    float* Ob = O2 + (size_t)bidx * C_I * DVOL;
#pragma unroll
    for (int i = 0; i < 8; ++i) {
      const int c = m0 + i + 8 * lh;
      const float v0 = a0[i] + bias2[c];
      const float v1 = a1[i] + bias2[c];
      Ob[(size_t)c * DVOL + p0 + lm]      = v0;
      Ob[(size_t)c * DVOL + p0 + 16 + lm] = v1;
      float s = v0 + v1, q = v0 * v0 + v1 * v1;
      s += __shfl_xor(s, 1, 32); s += __shfl_xor(s, 2, 32);
      s += __shfl_xor(s, 4, 32); s += __shfl_xor(s, 8, 32);
      q += __shfl_xor(q, 1, 32); q += __shfl_xor(q, 2, 32);
      q += __shfl_xor(q, 4, 32); q += __shfl_xor(q, 8, 32);
      if (lm == 0) {
        atomicAdd(&sum2[(size_t)(bidx * C_I + c) * NBUCK + bucket], s);
        atomicAdd(&sq2 [(size_t)(bidx * C_I + c) * NBUCK + bucket], q);
      }
    }
  }
}

// ---------------------------------------------------------------------------
// K3: reduce stat buckets -> per-(b,c) mean & rsqrt(var+eps).  256 threads.
// ---------------------------------------------------------------------------
__global__ __launch_bounds__(256)
void k_finalize(const float* __restrict__ sum, const float* __restrict__ sq,
                float* __restrict__ mean, float* __restrict__ rsq) {
  const int i = threadIdx.x;  // b*128 + c, 256 entries
  float s = 0.f, q = 0.f;
  for (int k = 0; k < NBUCK; ++k) {
    s += sum[(size_t)i * NBUCK + k];
    q += sq [(size_t)i * NBUCK + k];
  }
  const float m   = s * (1.0f / (float)DVOL);
  const float var = q * (1.0f / (float)DVOL) - m * m;
  mean[i] = m;
  rsq[i]  = rsqrtf(var + 1e-5f);
}

// ---------------------------------------------------------------------------
// K4: attention.  One thread = one position.  q normalized on the fly.
// Output written flat as (p,c) which IS the reference's reshape into (c,p).
// ---------------------------------------------------------------------------
__global__ __launch_bounds__(128)
void k_attn(const float* __restrict__ Q, const float* __restrict__ kb,
            const float* __restrict__ vb,
            const float* __restrict__ qmean, const float* __restrict__ qrsq,
            float* __restrict__ xflat) {
  __shared__ float kl[C_I][LLEN];
  __shared__ float vl[C_I][LLEN];
  __shared__ float ml[C_I], rl[C_I];
  const int tid  = threadIdx.x;
  const int blk  = blockIdx.x;
  const int bidx = blk / (DVOL / 128);
  const int p0   = (blk % (DVOL / 128)) * 128;
  for (int e = tid; e < C_I * LLEN; e += 128) {
    kl[e / LLEN][e % LLEN] = kb[bidx * C_I * LLEN + e];
    vl[e / LLEN][e % LLEN] = vb[bidx * C_I * LLEN + e];
  }
  if (tid < C_I) {
    ml[tid] = qmean[bidx * C_I + tid];
    rl[tid] = qrsq[bidx * C_I + tid];
  }
  __syncthreads();
  const float* Qb = Q + (size_t)bidx * C_I * DVOL;
  const int p = p0 + tid;
  float* dst = xflat + (size_t)bidx * C_I * DVOL + (size_t)p * C_I;
#pragma unroll 1
  for (int h = 0; h < 8; ++h) {
    float sim[LLEN];
#pragma unroll
    for (int l = 0; l < LLEN; ++l) sim[l] = 0.f;
#pragma unroll
    for (int j = 0; j < 16; ++j) {
      const int c = h * 16 + j;
      const float qv = (Qb[(size_t)c * DVOL + p] - ml[c]) * rl[c];
#pragma unroll
      for (int l = 0; l < LLEN; ++l) sim[l] += qv * kl[c][l];
    }
    float mx = -1e30f;
#pragma unroll
    for (int l = 0; l < LLEN; ++l) { sim[l] *= 0.25f; mx = fmaxf(mx, sim[l]); }
    float ss = 0.f;
#pragma unroll
    for (int l = 0; l < LLEN; ++l) { const float e = __expf(sim[l] - mx); sim[l] = e; ss += e; }
    const float inv = 1.0f / ss;
    float xo[16];
#pragma unroll
    for (int j = 0; j < 16; ++j) {
      const int c = h * 16 + j;
      float acc = 0.f;
#pragma unroll
      for (int l = 0; l < LLEN; ++l) acc += sim[l] * vl[c][l];
      xo[j] = acc * inv;
    }
    float4* d4 = (float4*)(dst + h * 16);
#pragma unroll
    for (int t = 0; t < 4; ++t)
      d4[t] = make_float4(xo[4 * t], xo[4 * t + 1], xo[4 * t + 2], xo[4 * t + 3]);
  }
}

// ---------------------------------------------------------------------------
// K7: fused  t = inorm(skip)*inorm(weight) -> relu(Wpo t) -> relu(Wg1 .) ->
// tanh(Wg2 .) -> out*gate.  Three chained WMMA GEMMs, 128 x 32 tile in LDS.
// ---------------------------------------------------------------------------
__global__ __launch_bounds__(256)
void k_final(const float* __restrict__ Sraw, const float* __restrict__ Wraw,
             const float* __restrict__ smean, const float* __restrict__ srsq,
             const float* __restrict__ wmean, const float* __restrict__ wrsq,
             const float* __restrict__ Wpo, const float* __restrict__ bpo,
             const float* __restrict__ Wg1, const float* __restrict__ bg1,
             const float* __restrict__ Wg2, const float* __restrict__ bg2,
             float* __restrict__ out) {
  __shared__ float ts[32][130];
  __shared__ float os[32][130];
  __shared__ float hs[32][130];
  __shared__ float bL[3 * C_I];
  const int tid  = threadIdx.x;
  const int blk  = blockIdx.x;
  const int bidx = blk / (DVOL / 32);
  const int p0   = (blk % (DVOL / 32)) * 32;
  const float* Sb = Sraw + (size_t)bidx * C_I * DVOL;
  const float* Wb = Wraw + (size_t)bidx * C_I * DVOL;

  if (tid < C_I) {
    bL[tid]           = bpo[tid];
    bL[C_I + tid]     = bg1[tid];
    bL[2 * C_I + tid] = bg2[tid];
  }
  for (int e = tid; e < C_I * 32; e += 256) {
    const int c = e >> 5, j = e & 31;
    const float sv = (Sb[(size_t)c * DVOL + p0 + j] - smean[bidx * C_I + c]) * srsq[bidx * C_I + c];
    const float wv = (Wb[(size_t)c * DVOL + p0 + j] - wmean[bidx * C_I + c]) * wrsq[bidx * C_I + c];
    ts[j][c] = sv * wv;
  }
  __syncthreads();

  const int wave = tid >> 5, lane = tid & 31;
  const int lm = lane & 15, lh = lane >> 4;
  const int m0 = wave * 16;

  {  // GEMM1: out = relu(Wpo @ t + bpo)
    v8f a0 = {0, 0, 0, 0, 0, 0, 0, 0}, a1 = {0, 0, 0, 0, 0, 0, 0, 0};
    gemm_tile_128(Wpo, m0, ts, lane, a0, a1);
#pragma unroll
    for (int i = 0; i < 8; ++i) {
      const int c = m0 + i + 8 * lh;
      os[lm][c]      = fmaxf(a0[i] + bL[c], 0.f);
      os[lm + 16][c] = fmaxf(a1[i] + bL[c], 0.f);
    }
  }
  __syncthreads();
  {  // GEMM2: h = relu(Wg1 @ out + bg1)
    v8f a0 = {0, 0, 0, 0, 0, 0, 0, 0}, a1 = {0, 0, 0, 0, 0, 0, 0, 0};
    gemm_tile_128(Wg1, m0, os, lane, a0, a1);
#pragma unroll
    for (int i = 0; i < 8; ++i) {
      const int c = m0 + i + 8 * lh;
      hs[lm][c]      = fmaxf(a0[i] + bL[C_I + c], 0.f);
      hs[lm + 16][c] = fmaxf(a1[i] + bL[C_I + c], 0.f);
    }
  }
  __syncthreads();
  {  // GEMM3: gate = tanh(Wg2 @ h + bg2);  result = out * gate
    v8f a0 = {0, 0, 0, 0, 0, 0, 0, 0}, a1 = {0, 0, 0, 0, 0, 0, 0, 0};
    gemm_tile_128(Wg2, m0, hs, lane, a0, a1);
    float* Ob = out + (size_t)bidx * C_I * DVOL;
#pragma unroll
    for (int i = 0; i < 8; ++i) {
      const int c = m0 + i + 8 * lh;
      const float g0 = tanhf(a0[i] + bL[2 * C_I + c]);
      const float g1v = tanhf(a1[i] + bL[2 * C_I + c]);
      Ob[(size_t)c * DVOL + p0 + lm]      = os[lm][c] * g0;
      Ob[(size_t)c * DVOL + p0 + 16 + lm] = os[lm + 16][c] * g1v;
    }
  }
}

// ---------------------------------------------------------------------------
extern "C" void kernel_launch(void* const* d_in, const int* in_sizes, int n_in,
                              void* d_out, int out_size, void* d_ws, size_t ws_size,
                              hipStream_t stream) {
  (void)in_sizes; (void)n_in; (void)out_size; (void)ws_size;
  const float* img  = (const float*)d_in[0];
  const float* lang = (const float*)d_in[1];
  const float* Wq  = (const float*)d_in[2];
  const float* bq  = (const float*)d_in[3];
  const float* Wk  = (const float*)d_in[4];
  const float* bk  = (const float*)d_in[5];
  const float* Wv  = (const float*)d_in[6];
  const float* bv  = (const float*)d_in[7];
  const float* Wpw = (const float*)d_in[8];
  const float* bpw = (const float*)d_in[9];
  const float* Wpi = (const float*)d_in[10];
  const float* bpi = (const float*)d_in[11];
  const float* Wpo = (const float*)d_in[12];
  const float* bpo = (const float*)d_in[13];
  const float* Wg1 = (const float*)d_in[14];
  const float* bg1 = (const float*)d_in[15];
  const float* Wg2 = (const float*)d_in[16];
  const float* bg2 = (const float*)d_in[17];
  float* out = (float*)d_out;

  const size_t NCD = (size_t)2 * C_I * DVOL;  // one full (b,c,p) tensor
  float* ws   = (float*)d_ws;
  float* bufA = ws;              // q_raw, later w_raw
  float* bufB = bufA + NCD;      // skip_raw
  float* bufC = bufB + NCD;      // x_flat (attention output)
  float* kbuf = bufC + NCD;
  float* vbuf = kbuf + 2 * C_I * LLEN;
  float* st   = vbuf + 2 * C_I * LLEN;
  float* sq_sum = st + 0 * 256 * NBUCK;
  float* sq_sq  = st + 1 * 256 * NBUCK;
  float* ss_sum = st + 2 * 256 * NBUCK;
  float* ss_sq  = st + 3 * 256 * NBUCK;
  float* sw_sum = st + 4 * 256 * NBUCK;
  float* sw_sq  = st + 5 * 256 * NBUCK;
  float* fin    = st + 6 * 256 * NBUCK;
  float* q_mean = fin;        float* q_rsq = fin + 256;
  float* s_mean = fin + 512;  float* s_rsq = fin + 768;
  float* w_mean = fin + 1024; float* w_rsq = fin + 1280;

  hipMemsetAsync(st, 0, (size_t)6 * 256 * NBUCK * sizeof(float), stream);

  k_kv<<<2, 256, 0, stream>>>(lang, Wk, bk, Wv, bv, kbuf, vbuf);

  k_gemm2<<<2 * (DVOL / 32), 256, 0, stream>>>(img,
      Wq, bq, bufA, sq_sum, sq_sq,
      Wpi, bpi, bufB, ss_sum, ss_sq);

  k_finalize<<<1, 256, 0, stream>>>(sq_sum, sq_sq, q_mean, q_rsq);
  k_finalize<<<1, 256, 0, stream>>>(ss_sum, ss_sq, s_mean, s_rsq);

  k_attn<<<2 * (DVOL / 128), 128, 0, stream>>>(bufA, kbuf, vbuf, q_mean, q_rsq, bufC);

  k_gemm2<<<2 * (DVOL / 32), 256, 0, stream>>>(bufC,
      Wpw, bpw, bufA, sw_sum, sw_sq,
      nullptr, nullptr, nullptr, nullptr, nullptr);

  k_finalize<<<1, 256, 0, stream>>>(sw_sum, sw_sq, w_mean, w_rsq);

  k_final<<<2 * (DVOL / 32), 256, 0, stream>>>(bufB, bufA,
      s_mean, s_rsq, w_mean, w_rsq,
      Wpo, bpo, Wg1, bg1, Wg2, bg2, out);
}